// LateralMovementGNN_84817014161825
// MI455X (gfx1250) — compile-verified
//
#include <hip/hip_runtime.h>

#define N_NODES 50000
#define N_EDGES 800000

typedef float v2f __attribute__((ext_vector_type(2)));
typedef float v8f __attribute__((ext_vector_type(8)));

// ---------------- degree / inverse degree ----------------

__global__ void deg_kernel(const int* __restrict__ dst, float* __restrict__ deg) {
  int e = blockIdx.x * blockDim.x + threadIdx.x;
  if (e < N_EDGES) atomicAdd(&deg[dst[e]], 1.0f);
}

__global__ void invdeg_kernel(float* __restrict__ deg) {
  int i = blockIdx.x * blockDim.x + threadIdx.x;
  if (i < N_NODES) {
    float d = deg[i];
    deg[i] = d > 0.0f ? 1.0f / d : 0.0f;
  }
}

// ---------------- edge scatter-add (mean aggregation numerator) ----------------
// One thread per float4 chunk of one edge: b128 gather + 4 f32 atomics.
// agg (12.8 MB) fits in the 192 MB L2, so atomics resolve at L2 atomic units.

template<int F>
__global__ void scatter_add_kernel(const float* __restrict__ feat,
                                   const int* __restrict__ src,
                                   const int* __restrict__ dst,
                                   float* __restrict__ agg) {
  constexpr int PER_EDGE = F / 4;
  long tid = (long)blockIdx.x * blockDim.x + threadIdx.x;
  int e = (int)(tid / PER_EDGE);
  if (e >= N_EDGES) return;
  int f0 = (int)(tid % PER_EDGE) * 4;
  int s = src[e];
  int d = dst[e];
  const float4 v = *(const float4*)(feat + (size_t)s * F + f0);
  float* p = agg + (size_t)d * F + f0;
  atomicAdd(p + 0, v.x);
  atomicAdd(p + 1, v.y);
  atomicAdd(p + 2, v.z);
  atomicAdd(p + 3, v.w);
}

// ---------------- SAGE layer: out = (agg * inv_deg) @ Wl + b + hin @ Wr ----------------
// One wave32 owns a 16-node tile. V_WMMA_F32_16X16X4_F32, K-outer loop.
// Each A k-slice is loaded ONCE and fed to 2*NT WMMAs (NT independent
// accumulator chains -> no WMMA->WMMA RAW hazards, A traffic / NT).
// A layout (16x4 f32): lane L holds M = L%16, K = 2*(L/16) .. +1  (2 VGPRs)
// B layout (4x16 f32): lane L holds N = L%16, K = 2*(L/16) .. +1
// C/D (16x16 f32):     reg r, lane L -> [M = r + 8*(L/16)][N = L%16]

template<int K, int NT, int DSTORE, bool RELU>
__global__ void sage_layer_kernel(const float* __restrict__ hin,     // [N, K]
                                  const float* __restrict__ agg,     // [N, K]
                                  const float* __restrict__ invdeg,  // [N]
                                  const float* __restrict__ Wl,      // [K, DSTORE]
                                  const float* __restrict__ bias,    // [DSTORE]
                                  const float* __restrict__ Wr,      // [K, DSTORE]
                                  float* __restrict__ out) {         // [N, DSTORE]
  constexpr int TILES_TOTAL = N_NODES / 16;      // 3125 exactly
  constexpr bool FULL = (NT * 16 == DSTORE);     // no N padding needed
  int wave = (int)((blockIdx.x * blockDim.x + threadIdx.x) >> 5);
  int lane = threadIdx.x & 31;
  if (wave >= TILES_TOTAL) return;               // wave-uniform: EXEC stays all-ones

  int m = lane & 15;        // A-row / B-col / C-col index
  int khalf = lane >> 4;    // which K pair this lane holds
  int row_base = wave * 16;
  int arow = row_base + m;
  float inv = invdeg[arow];

  // bias broadcast: C[m][n] init = b[n]
  v8f c[NT];
#pragma unroll
  for (int nt = 0; nt < NT; ++nt) {
    int n = nt * 16 + m;
    float bv = (FULL || n < DSTORE) ? bias[n] : 0.0f;
#pragma unroll
    for (int r = 0; r < 8; ++r) c[nt][r] = bv;
  }

#pragma unroll
  for (int k0 = 0; k0 < K; k0 += 4) {
    int ka = k0 + 2 * khalf;
    v2f a1, a2;
    const float* ap = agg + (size_t)arow * K + ka;
    a1.x = ap[0] * inv;                          // fold mean-scaling into A
    a1.y = ap[1] * inv;
    const float* hp = hin + (size_t)arow * K + ka;
    a2.x = hp[0];
    a2.y = hp[1];

#pragma unroll
    for (int nt = 0; nt < NT; ++nt) {
      int n = nt * 16 + m;
      v2f bl, br;
      if (FULL || n < DSTORE) {
        bl.x = Wl[(size_t)ka * DSTORE + n];
        bl.y = Wl[(size_t)(ka + 1) * DSTORE + n];
        br.x = Wr[(size_t)ka * DSTORE + n];
        br.y = Wr[(size_t)(ka + 1) * DSTORE + n];
      } else {
        bl.x = 0.0f; bl.y = 0.0f;
        br.x = 0.0f; br.y = 0.0f;
      }
      // D = A*B + C, full-precision f32 WMMA
      c[nt] = __builtin_amdgcn_wmma_f32_16x16x4_f32(false, a1, false, bl, (short)0,
                                                    c[nt], false, false);
      c[nt] = __builtin_amdgcn_wmma_f32_16x16x4_f32(false, a2, false, br, (short)0,
                                                    c[nt], false, false);
    }
  }

#pragma unroll
  for (int nt = 0; nt < NT; ++nt) {
    int n = nt * 16 + m;
    if (FULL || n < DSTORE) {
#pragma unroll
      for (int r = 0; r < 8; ++r) {
        int orow = row_base + r + 8 * khalf;
        float v = c[nt][r];
        if (RELU) v = fmaxf(v, 0.0f);
        out[(size_t)orow * DSTORE + n] = v;
      }
    }
  }
}

// ---------------- log_softmax over 2 classes, in place ----------------

__global__ void logsoftmax2_kernel(float* __restrict__ out) {
  int i = blockIdx.x * blockDim.x + threadIdx.x;
  if (i >= N_NODES) return;
  float a = out[2 * i];
  float b = out[2 * i + 1];
  float mx = fmaxf(a, b);
  float lse = mx + logf(expf(a - mx) + expf(b - mx));
  out[2 * i] = a - lse;
  out[2 * i + 1] = b - lse;
}

// ---------------- launch ----------------

extern "C" void kernel_launch(void* const* d_in, const int* in_sizes, int n_in,
                              void* d_out, int out_size, void* d_ws, size_t ws_size,
                              hipStream_t stream) {
  const float* x   = (const float*)d_in[0];
  const int*   ei  = (const int*)d_in[1];   // [2, N_EDGES] int32 (JAX x64 disabled)
  const float* W1l = (const float*)d_in[2];
  const float* b1  = (const float*)d_in[3];
  const float* W1r = (const float*)d_in[4];
  const float* W2l = (const float*)d_in[5];
  const float* b2  = (const float*)d_in[6];
  const float* W2r = (const float*)d_in[7];
  const float* W3l = (const float*)d_in[8];
  const float* b3  = (const float*)d_in[9];
  const float* W3r = (const float*)d_in[10];
  float* out = (float*)d_out;

  const int* src = ei;
  const int* dst = ei + N_EDGES;

  // workspace layout (floats): deg[50048] | agg[50000*64] | hA[50000*64] | hB[50000*64]
  float* deg = (float*)d_ws;
  float* agg = deg + 50048;
  float* hA  = agg + (size_t)N_NODES * 64;
  float* hB  = hA + (size_t)N_NODES * 64;

  // degrees -> inverse degrees (in place)
  hipMemsetAsync(deg, 0, N_NODES * sizeof(float), stream);
  deg_kernel<<<(N_EDGES + 255) / 256, 256, 0, stream>>>(dst, deg);
  invdeg_kernel<<<(N_NODES + 255) / 256, 256, 0, stream>>>(deg);

  const int LBLOCKS = (N_NODES / 16 + 7) / 8;  // 8 waves per 256-thread block

  // ---- layer 1 (K=16 -> 64, relu) ----
  hipMemsetAsync(agg, 0, (size_t)N_NODES * 16 * sizeof(float), stream);
  scatter_add_kernel<16><<<(int)(((long)N_EDGES * 4 + 255) / 256), 256, 0, stream>>>(
      x, src, dst, agg);
  sage_layer_kernel<16, 4, 64, true><<<LBLOCKS, 256, 0, stream>>>(
      x, agg, deg, W1l, b1, W1r, hA);

  // ---- layer 2 (K=64 -> 64, relu) ----
  hipMemsetAsync(agg, 0, (size_t)N_NODES * 64 * sizeof(float), stream);
  scatter_add_kernel<64><<<(int)(((long)N_EDGES * 16 + 255) / 256), 256, 0, stream>>>(
      hA, src, dst, agg);
  sage_layer_kernel<64, 4, 64, true><<<LBLOCKS, 256, 0, stream>>>(
      hA, agg, deg, W2l, b2, W2r, hB);

  // ---- layer 3 (K=64 -> 2, no relu; N padded 2 -> 16) ----
  hipMemsetAsync(agg, 0, (size_t)N_NODES * 64 * sizeof(float), stream);
  scatter_add_kernel<64><<<(int)(((long)N_EDGES * 16 + 255) / 256), 256, 0, stream>>>(
      hB, src, dst, agg);
  sage_layer_kernel<64, 1, 2, false><<<LBLOCKS, 256, 0, stream>>>(
      hB, agg, deg, W3l, b3, W3r, out);

  // ---- log_softmax ----
  logsoftmax2_kernel<<<(N_NODES + 255) / 256, 256, 0, stream>>>(out);
}